// Linear_12541304504952
// MI455X (gfx1250) — compile-verified
//
#include <hip/hip_runtime.h>

typedef __attribute__((ext_vector_type(2))) float v2f;
typedef __attribute__((ext_vector_type(4))) float v4f;
typedef __attribute__((ext_vector_type(8))) float v8f;
typedef int v4i_vs __attribute__((vector_size(4 * sizeof(int))));

#define AS1 __attribute__((address_space(1)))
#define AS3 __attribute__((address_space(3)))

#define IN_DIM   128
#define OUT_DIM  128
#define SROWS    64            // rows of x per stripe
#define LDW      132           // padded LDS row stride (floats): conflict-free frag reads
#define NBLOCKS  2048          // persistent workgroups

#define W_FLOATS    (OUT_DIM * LDW)               // 16896 floats (66 KB)
#define XBUF_FLOATS (SROWS * LDW)                 //  8448 floats (33 KB)
#define SMEM_FLOATS (W_FLOATS + 2 * XBUF_FLOATS)  // 33792 floats = 132 KB of 320 KB/WGP

#if __has_builtin(__builtin_amdgcn_global_load_async_to_lds_b128) && \
    __has_builtin(__builtin_amdgcn_s_wait_asynccnt)
#define USE_ASYNC 1
#define WAIT_ASYNC(n) __builtin_amdgcn_s_wait_asynccnt(n)
#else
#define USE_ASYNC 0
#define WAIT_ASYNC(n) do { } while (0)
#endif

__device__ __forceinline__ void copy16(const float* __restrict__ g, float* l) {
#if USE_ASYNC
    // async DMA global -> LDS, tracked by ASYNCcnt (no VGPR round trip)
    __builtin_amdgcn_global_load_async_to_lds_b128(
        (AS1 v4i_vs*)g, (AS3 v4i_vs*)l, 0, 0);
#else
    *(v4f*)l = *(const v4f*)g;
#endif
}

// 128x128 W into padded LDS: 4096 float4 chunks, 16 per thread
__device__ __forceinline__ void stage_w(const float* __restrict__ w, float* lds_w, int tid) {
    #pragma unroll
    for (int j = 0; j < 16; ++j) {
        const int c = tid + j * 256;
        const int row = c >> 5, col = (c & 31) * 4;
        copy16(w + row * IN_DIM + col, lds_w + row * LDW + col);
    }
}

// 64x128 x-stripe into padded LDS: 2048 float4 chunks, 8 per thread
__device__ __forceinline__ void stage_x(const float* __restrict__ x, float* xbuf,
                                        long long stripe, int tid) {
    const float* src = x + stripe * (long long)(SROWS * IN_DIM);
    #pragma unroll
    for (int j = 0; j < 8; ++j) {
        const int c = tid + j * 256;
        const int row = c >> 5, col = (c & 31) * 4;
        copy16(src + row * IN_DIM + col, xbuf + row * LDW + col);
    }
}

__global__ __launch_bounds__(256) void linear_wmma_pipe(
    const float* __restrict__ x,      // [B, 128]
    const float* __restrict__ w,      // [128, 128] (OUT, IN) row-major
    const float* __restrict__ bias,   // [128]
    float* __restrict__ y,            // [B, 128]
    int nstripes)
{
    extern __shared__ float smem[];
    float* lds_w = smem;
    float* xbuf0 = smem + W_FLOATS;
    float* xbuf1 = xbuf0 + XBUF_FLOATS;

    const int tid  = threadIdx.x;
    const int lane = tid & 31;
    const int wave = tid >> 5;
    const int half = lane >> 4;   // 0: lanes 0-15, 1: lanes 16-31
    const int lrow = lane & 15;
    const int mt   = wave & 3;    // this wave's 16-row M tile within the stripe
    const int nh   = wave >> 2;   // this wave's 64-column N half

    // ---- prologue: W (once per persistent block) + first stripe into buf0 ----
    stage_w(w, lds_w, tid);
    int s = blockIdx.x;
    if (s < nstripes) stage_x(x, xbuf0, s, tid);
    WAIT_ASYNC(0);
    __syncthreads();

    int ibuf = 0;
    for (; s < nstripes; s += (int)gridDim.x) {
        const int snext = s + (int)gridDim.x;
        if (snext < nstripes) {
            // prefetch next stripe into the spare buffer while we compute this one
            stage_x(x, ibuf ? xbuf0 : xbuf1, snext, tid);
            WAIT_ASYNC(8);   // in-order: <=8 outstanding => previous stripe landed
        } else {
            WAIT_ASYNC(0);   // nothing new issued: drain current stripe's loads
        }
        __syncthreads();

        const float* xb = ibuf ? xbuf1 : xbuf0;

        // A fragments (16x4 fp32): lane<16 -> K=k0,k0+1 ; lane>=16 -> K=k0+2,k0+3
        v2f a[32];
        #pragma unroll
        for (int ks = 0; ks < 32; ++ks)
            a[ks] = *(const v2f*)(xb + (mt * 16 + lrow) * LDW + 4 * ks + 2 * half);

        #pragma unroll
        for (int j = 0; j < 4; ++j) {
            const int n0 = (nh * 4 + j) * 16;
            const float bv = bias[n0 + lrow];   // C layout: N = lane%16
            v8f c;
            #pragma unroll
            for (int v = 0; v < 8; ++v) c[v] = bv;

            #pragma unroll
            for (int ks = 0; ks < 32; ++ks) {
                // B fragment (4x16): B[k][n] = W[n][k]
                v2f b = *(const v2f*)(lds_w + (n0 + lrow) * LDW + 4 * ks + 2 * half);
                c = __builtin_amdgcn_wmma_f32_16x16x4_f32(false, a[ks], false, b,
                                                          (short)0, c, false, false);
            }

            const long long rbase = (long long)s * SROWS + mt * 16 + 8 * half;
            #pragma unroll
            for (int v = 0; v < 8; ++v)
                y[(rbase + v) * OUT_DIM + n0 + lrow] = c[v];
        }

        __syncthreads();   // WAR: everyone done with xb before it is refilled
        ibuf ^= 1;
    }
}

extern "C" void kernel_launch(void* const* d_in, const int* in_sizes, int n_in,
                              void* d_out, int out_size, void* d_ws, size_t ws_size,
                              hipStream_t stream) {
    const float* x = (const float*)d_in[0];
    const float* w = (const float*)d_in[1];
    const float* b = (const float*)d_in[2];
    float* y = (float*)d_out;

    const int rows = in_sizes[0] / IN_DIM;   // 524288
    const int nstripes = rows / SROWS;       // 8192
    int grid = NBLOCKS;                      // persistent: 4 stripes per block
    if (grid > nstripes) grid = nstripes;
    const size_t shmem = (size_t)SMEM_FLOATS * sizeof(float);

    linear_wmma_pipe<<<grid, 256, shmem, stream>>>(x, w, b, y, nstripes);
}